// encoder_3959959847164
// MI455X (gfx1250) — compile-verified
//
#include <hip/hip_runtime.h>
#include <hip/hip_bf16.h>

// Problem constants (from reference): B=256, T=127, K=128, H=256
#define PB   256
#define PT   127
#define PK   128
#define PH   256
#define P4H  1024          // 4*H
#define PKA  384           // K + H (concatenated GEMM reduction dim)

typedef float v2f  __attribute__((ext_vector_type(2)));
typedef float v8f  __attribute__((ext_vector_type(8)));
typedef unsigned int u32x4 __attribute__((ext_vector_type(4)));
typedef int   i32x4 __attribute__((ext_vector_type(4)));
typedef int   i32x8 __attribute__((ext_vector_type(8)));

__device__ __forceinline__ float sigm(float x) { return 1.0f / (1.0f + expf(-x)); }

// ---------------------------------------------------------------------------
// TDM descriptor builders (CDNA5 ISA)
// ---------------------------------------------------------------------------
// Group 0: [1:0]=count=1, [63:32]=lds_addr, [120:64]=global_addr, [127:126]=type=2
__device__ __forceinline__ u32x4 tdm_g0(unsigned lds_addr, const void* gptr) {
    unsigned long long ga = (unsigned long long)(uintptr_t)gptr;
    u32x4 g;
    g.x = 1u;
    g.y = lds_addr;
    g.z = (unsigned)(ga & 0xFFFFFFFFu);
    g.w = (unsigned)((ga >> 32) & 0x01FFFFFFu) | (2u << 30);
    return g;
}

// Group 1: [17:16]=data_size, [20]=pad_enable, [24:22]=pad_interval, [31:25]=pad_amount,
// [79:48]=tensor_dim0, [111:80]=tensor_dim1, [127:112]=tile_dim0, [143:128]=tile_dim1,
// [207:160]=tensor_dim0_stride
__device__ __forceinline__ i32x8 tdm_g1(unsigned flags, unsigned td0, unsigned td1,
                                        unsigned tile0, unsigned tile1,
                                        unsigned long long stride0) {
    i32x8 g;
    g[0] = (int)flags;
    g[1] = (int)((td0 & 0xFFFFu) << 16);
    g[2] = (int)((td0 >> 16) | ((td1 & 0xFFFFu) << 16));
    g[3] = (int)((td1 >> 16) | ((tile0 & 0xFFFFu) << 16));
    g[4] = (int)(tile1 & 0xFFFFu);               // tile_dim1; tile_dim2 = 0
    g[5] = (int)(stride0 & 0xFFFFFFFFull);
    g[6] = (int)((stride0 >> 32) & 0xFFFFull);   // tensor_dim1_stride = 0
    g[7] = 0;
    return g;
}

#define TDM_F32          (2u << 16)                               // data_size = 4 bytes
#define TDM_F32_PAD33    ((2u << 16) | (1u << 20) | (4u << 22))   // +1 DWORD pad / 32 DWORDs

// ---------------------------------------------------------------------------
// Setup kernels
// ---------------------------------------------------------------------------
__global__ void k_zero_state(float* __restrict__ Acat, float* __restrict__ cbuf) {
    int idx = blockIdx.x * blockDim.x + threadIdx.x;
    if (idx < PB * PKA) Acat[idx] = 0.0f;
    int i2 = idx - PB * PKA;
    if (i2 >= 0 && i2 < PB * PH) cbuf[i2] = 0.0f;
}

// x_score[b,k] = sum_t input[b,t,k] * W_x[t] + fc_b   (W_x = fc_w[2H:])
__global__ __launch_bounds__(PK) void k_xscore(const float* __restrict__ in,
                                               const float* __restrict__ fcw,
                                               const float* __restrict__ fcb,
                                               float* __restrict__ xscore) {
    int b = blockIdx.x;
    int k = threadIdx.x;
    float s = fcb[0];
    for (int t = 0; t < PT; ++t)
        s += in[(b * PT + t) * PK + k] * fcw[2 * PH + t];
    xscore[b * PK + k] = s;
}

// W_cat[kk][j] (K-major, 384x1024): kk<128 -> W_ih[j][kk], else W_hh[j][kk-128]
__global__ void k_packW(const float* __restrict__ Wih, const float* __restrict__ Whh,
                        float* __restrict__ Wcat) {
    int idx = blockIdx.x * blockDim.x + threadIdx.x;
    if (idx >= PKA * P4H) return;
    int kk = idx >> 10;
    int j  = idx & (P4H - 1);
    Wcat[idx] = (kk < PK) ? Wih[j * PK + kk] : Whh[j * PH + (kk - PK)];
}

// ---------------------------------------------------------------------------
// Phase 1 (per step): scalar score -> softmax over K -> x = alpha * x_t
// ---------------------------------------------------------------------------
__global__ __launch_bounds__(PK) void k_attn(const float* __restrict__ in,
                                             const float* __restrict__ fcw,
                                             const float* __restrict__ xscore,
                                             float* __restrict__ Acat,
                                             const float* __restrict__ cbuf,
                                             float* __restrict__ wout, int t) {
    __shared__ float red[PK];
    const int b = blockIdx.x;
    const int k = threadIdx.x;
    const float* hrow = Acat + b * PKA + PK;   // h lives in cols [128,384)
    const float* crow = cbuf + b * PH;

    float p = hrow[k] * fcw[k] + hrow[PK + k] * fcw[PK + k]
            + crow[k] * fcw[PH + k] + crow[PK + k] * fcw[PH + PK + k];
    red[k] = p; __syncthreads();
    for (int s = 64; s > 0; s >>= 1) { if (k < s) red[k] += red[k + s]; __syncthreads(); }
    const float sbc = red[0];
    __syncthreads();

    float sc = xscore[b * PK + k] + sbc;
    red[k] = sc; __syncthreads();
    for (int s = 64; s > 0; s >>= 1) { if (k < s) red[k] = fmaxf(red[k], red[k + s]); __syncthreads(); }
    const float mx = red[0];
    __syncthreads();

    float e = expf(sc - mx);
    red[k] = e; __syncthreads();
    for (int s = 64; s > 0; s >>= 1) { if (k < s) red[k] += red[k + s]; __syncthreads(); }
    const float denom = red[0];

    float alpha = e / denom;
    float x = alpha * in[(b * PT + t) * PK + k];
    Acat[b * PKA + k] = x;
    wout[(b * PT + t) * PK + k] = x;
}

// ---------------------------------------------------------------------------
// Phase 2 (per step): gates = A_cat(256x384) @ W_cat(384x1024), f32 WMMA.
// Tile staging via the Tensor Data Mover, double-buffered in LDS: wave 0
// issues tensor_load_to_lds for chunk k+1 while all waves run the WMMAs of
// chunk k; s_wait_tensorcnt + workgroup barrier flips the buffer.
// TDM pad feature (+1 DWORD every 32) reproduces the pitch-33 A layout.
// ---------------------------------------------------------------------------
#define TILE_M 32
#define TILE_N 64
#define KC     32
#define APITCH 33
#define NCHUNK (PKA / KC)   // 12

__global__ __launch_bounds__(256) void k_gemm(const float* __restrict__ Acat,
                                              const float* __restrict__ Wcat,
                                              float* __restrict__ gates) {
    __shared__ float ldsA[2][TILE_M * APITCH];
    __shared__ float ldsB[2][KC * TILE_N];

    const int tid  = threadIdx.x;
    const int r0   = blockIdx.x * TILE_M;   // 8 M-tiles
    const int n0   = blockIdx.y * TILE_N;   // 16 N-tiles
    const int wave = tid >> 5;
    const int lane = tid & 31;
    const int half = lane >> 4;
    const int lr   = lane & 15;
    const int wm   = wave >> 2;             // 0..1  (M sub-tile)
    const int wn   = wave & 3;              // 0..3  (N sub-tile)

    // Constant descriptor group-1s (uniform -> SGPRs)
    const i32x8 g1A = tdm_g1(TDM_F32_PAD33, PKA, PB, KC, TILE_M, PKA);
    const i32x8 g1B = tdm_g1(TDM_F32, P4H, PKA, TILE_N, KC, P4H);
    const i32x4 gz4 = {};
    const i32x8 gz8 = {};

    // Prime buffer 0 with chunk 0 (wave 0 drives the TDM)
    if (tid < 32) {
        __builtin_amdgcn_tensor_load_to_lds(
            tdm_g0((unsigned)(uintptr_t)(void*)&ldsA[0][0], &Acat[r0 * PKA + 0]),
            g1A, gz4, gz4, gz8, 0);
        __builtin_amdgcn_tensor_load_to_lds(
            tdm_g0((unsigned)(uintptr_t)(void*)&ldsB[0][0], &Wcat[0 * P4H + n0]),
            g1B, gz4, gz4, gz8, 0);
        __builtin_amdgcn_s_wait_tensorcnt(0);
    }
    __syncthreads();

    v8f acc = {};

    for (int ci = 0; ci < NCHUNK; ++ci) {
        const int buf = ci & 1;
        // Kick off the next chunk's DMA into the other buffer
        if (ci + 1 < NCHUNK && tid < 32) {
            const int kc = (ci + 1) * KC;
            __builtin_amdgcn_tensor_load_to_lds(
                tdm_g0((unsigned)(uintptr_t)(void*)&ldsA[buf ^ 1][0],
                       &Acat[r0 * PKA + kc]),
                g1A, gz4, gz4, gz8, 0);
            __builtin_amdgcn_tensor_load_to_lds(
                tdm_g0((unsigned)(uintptr_t)(void*)&ldsB[buf ^ 1][0],
                       &Wcat[kc * P4H + n0]),
                g1B, gz4, gz4, gz8, 0);
        }

        // 8 x V_WMMA_F32_16X16X4_F32 over this chunk
        #pragma unroll
        for (int kk = 0; kk < KC; kk += 4) {
            v2f a, bf;
            const int arow = wm * 16 + lr;
            a.x = ldsA[buf][arow * APITCH + kk + half * 2];
            a.y = ldsA[buf][arow * APITCH + kk + half * 2 + 1];
            const int bcol = wn * 16 + lr;
            bf.x = ldsB[buf][(kk + half * 2) * TILE_N + bcol];
            bf.y = ldsB[buf][(kk + half * 2 + 1) * TILE_N + bcol];
            acc = __builtin_amdgcn_wmma_f32_16x16x4_f32(
                false, a, false, bf, (short)0, acc, false, false);
        }

        // Next buffer must be landed (issuing wave) and this one fully read (all)
        if (ci + 1 < NCHUNK && tid < 32)
            __builtin_amdgcn_s_wait_tensorcnt(0);
        __syncthreads();
    }

    // C/D layout: VGPR i -> (M=i, N=lane) lanes 0-15; (M=8+i, N=lane-16) lanes 16-31
    const int col   = n0 + wn * 16 + lr;
    const int rbase = r0 + wm * 16 + half * 8;
    #pragma unroll
    for (int i = 0; i < 8; ++i)
        gates[(rbase + i) * P4H + col] = acc[i];
}

// ---------------------------------------------------------------------------
// Phase 3 (per step): bias + LSTM cell update; write h to output and A_cat.
// ---------------------------------------------------------------------------
__global__ __launch_bounds__(PH) void k_cell(const float* __restrict__ gates,
                                             const float* __restrict__ bih,
                                             const float* __restrict__ bhh,
                                             float* __restrict__ cbuf,
                                             float* __restrict__ Acat,
                                             float* __restrict__ eout, int t) {
    int b = blockIdx.x, j = threadIdx.x;
    const float* g = gates + b * P4H;
    float gi = g[j]           + bih[j]           + bhh[j];
    float gf = g[PH + j]      + bih[PH + j]      + bhh[PH + j];
    float gg = g[2 * PH + j]  + bih[2 * PH + j]  + bhh[2 * PH + j];
    float go = g[3 * PH + j]  + bih[3 * PH + j]  + bhh[3 * PH + j];
    float co = cbuf[b * PH + j];
    float cn = sigm(gf) * co + sigm(gi) * tanhf(gg);
    float hn = sigm(go) * tanhf(cn);
    cbuf[b * PH + j] = cn;
    Acat[b * PKA + PK + j] = hn;
    eout[(b * PT + t) * PH + j] = hn;
}

// ---------------------------------------------------------------------------
extern "C" void kernel_launch(void* const* d_in, const int* in_sizes, int n_in,
                              void* d_out, int out_size, void* d_ws, size_t ws_size,
                              hipStream_t stream) {
    const float* in  = (const float*)d_in[0];   // (B,T,K)
    const float* Wih = (const float*)d_in[1];   // (4H,K)
    const float* Whh = (const float*)d_in[2];   // (4H,H)
    const float* bih = (const float*)d_in[3];   // (4H,)
    const float* bhh = (const float*)d_in[4];   // (4H,)
    const float* fcw = (const float*)d_in[5];   // (1, 2H+T)
    const float* fcb = (const float*)d_in[6];   // (1,)

    float* out  = (float*)d_out;
    float* wout = out;                          // weight:        B*T*K
    float* eout = out + (size_t)PB * PT * PK;   // encode_weight: B*T*H

    float* ws     = (float*)d_ws;
    float* Acat   = ws;                         // B x 384 ([x | h])
    float* cbuf   = Acat   + (size_t)PB * PKA;  // B x 256
    float* xscore = cbuf   + (size_t)PB * PH;   // B x 128
    float* Wcat   = xscore + (size_t)PB * PK;   // 384 x 1024
    float* gates  = Wcat   + (size_t)PKA * P4H; // B x 1024

    k_zero_state<<<(PB * PKA + PB * PH + 255) / 256, 256, 0, stream>>>(Acat, cbuf);
    k_packW<<<(PKA * P4H + 255) / 256, 256, 0, stream>>>(Wih, Whh, Wcat);
    k_xscore<<<PB, PK, 0, stream>>>(in, fcw, fcb, xscore);

    dim3 gGemm(PB / TILE_M, P4H / TILE_N);      // (8, 16)
    for (int t = 0; t < PT; ++t) {
        k_attn<<<PB, PK, 0, stream>>>(in, fcw, xscore, Acat, cbuf, wout, t);
        k_gemm<<<gGemm, 256, 0, stream>>>(Acat, Wcat, gates);
        k_cell<<<PB, PH, 0, stream>>>(gates, bih, bhh, cbuf, Acat, eout, t);
    }
}